// GC_MC_14113262535118
// MI455X (gfx1250) — compile-verified
//
#include <hip/hip_runtime.h>

#define NUM_USERS 200000
#define NUM_ITEMS 100000
#define NUM_NODES 300000
#define NUM_EDGES 600000
#define DIM 128
#define BATCH 16384
#define NSLOTS (2 * BATCH)
#define WSTRIDE 132   // padded LDS row stride (floats) -> conflict-free WMMA reads

typedef __attribute__((ext_vector_type(2))) float v2f;
typedef __attribute__((ext_vector_type(4))) float v4f;
typedef __attribute__((ext_vector_type(8))) float v8f;

// ---- workspace layout (bytes) ----
#define WS_SLOT_OFF   0u
#define WS_AGG_OFF    1200000u                 // 300000*4, 128B aligned
#define WS_CNT_OFF    17977216u                // + 32768*128*4, 128B aligned
#define WS_WL_OFF     17977344u                // + 128B pad; worklist int2[600000]

// ---------------- kernel 1: slot table init (+ zero worklist counter) ------
__global__ void k_init_slots(int* __restrict__ slot, unsigned* __restrict__ gcount) {
    int i = blockIdx.x * blockDim.x + threadIdx.x;
    if (i < NUM_NODES) slot[i] = -1;
    if (i == 0) *gcount = 0;
}

// ---------------- kernel 2: mark needed nodes + zero compact agg rows ------
// one wave per batch slot (32768 waves); duplicates race on slot[node] with
// plain stores -> any single winner is consistent within this call.
__global__ void k_mark_zero(const int* __restrict__ ui, const int* __restrict__ ii,
                            int* __restrict__ slot, float* __restrict__ agg) {
    int w    = (blockIdx.x * blockDim.x + threadIdx.x) >> 5;
    int lane = threadIdx.x & 31;
    int node = (w < BATCH) ? ui[w] : (NUM_USERS + ii[w - BATCH]);
    if (lane == 0) slot[node] = w;
    v4f z = {};
    *(v4f*)(agg + (size_t)w * DIM + lane * 4) = z;
}

// ---------------- kernel 3a: edge filter + stream compaction ----------------
// thread per edge: coalesced dst/slot scan; LDS-aggregated counter, one
// global atomic per block; emits (slot_row, src_node) hits to worklist.
__global__ void k_compact(const int* __restrict__ eidx, const int* __restrict__ slot,
                          unsigned* __restrict__ gcount, int2* __restrict__ wl) {
    __shared__ unsigned lcount, lbase;
    int tid = threadIdx.x;
    if (tid == 0) lcount = 0;
    __syncthreads();

    int e = blockIdx.x * blockDim.x + tid;
    int srow = -1, s = 0;
    if (e < NUM_EDGES) {
        srow = slot[eidx[NUM_EDGES + e]];     // dst row of edge_index
        if (srow >= 0) s = eidx[e];           // src row of edge_index
    }
    bool hit = (srow >= 0);
    unsigned my = 0;
    if (hit) my = atomicAdd(&lcount, 1u);     // ds_add_rtn_u32
    __syncthreads();
    if (tid == 0) lbase = atomicAdd(gcount, lcount);
    __syncthreads();
    if (hit) wl[lbase + my] = make_int2(srow, s);
}

// ---------------- kernel 3b: persistent apply (wave per worklist entry) ----
__global__ void k_apply(const int2* __restrict__ wl, const unsigned* __restrict__ gcount,
                        const float* __restrict__ uemb, const float* __restrict__ iemb,
                        float* __restrict__ agg) {
    int lane = threadIdx.x & 31;
    unsigned wave = (blockIdx.x * blockDim.x + threadIdx.x) >> 5;
    unsigned nw   = (gridDim.x * blockDim.x) >> 5;
    unsigned count = *gcount;
    for (unsigned j = wave; j < count; j += nw) {
        int2 t = wl[j];
        const float* nrow = (t.y < NUM_USERS)
                                ? (uemb + (size_t)t.y * DIM)
                                : (iemb + (size_t)(t.y - NUM_USERS) * DIM);
        float* arow = agg + (size_t)t.x * DIM;
#pragma unroll
        for (int k = 0; k < 4; ++k)
            unsafeAtomicAdd(arow + lane + 32 * k, nrow[lane + 32 * k]);
    }
}

// ---------------- kernel 4: WMMA GEMM (prop = agg @ W^T + b) fused with dot --
// 8 waves/block, each wave owns 16 batch rows; full-wave EXEC for WMMA.
__global__ void __launch_bounds__(256)
k_gemm_dot(const int* __restrict__ ui, const int* __restrict__ ii,
           const int* __restrict__ slot, const float* __restrict__ W,
           const float* __restrict__ bias, const float* __restrict__ agg,
           float* __restrict__ out) {
    __shared__ float ldsW[16 * WSTRIDE];      // one 16-row W tile, padded

    int tid  = threadIdx.x;
    int wave = (blockIdx.x * 256 + tid) >> 5;
    int lane = tid & 31;
    int m    = lane & 15;                     // row within half (M for A, N for B)
    int h    = lane >> 4;                     // half-wave id
    int row_base = wave * 16;

    int ur = slot[ui[row_base + m]];
    int ir = slot[NUM_USERS + ii[row_base + m]];
    const float* uap = agg + (size_t)ur * DIM + 2 * h;
    const float* iap = agg + (size_t)ir * DIM + 2 * h;

    float partial[8];
#pragma unroll
    for (int v = 0; v < 8; ++v) partial[v] = 0.0f;

    for (int nt = 0; nt < 8; ++nt) {
        // stage W rows [nt*16, nt*16+16) into padded LDS via 16B transfers
        __syncthreads();
#pragma unroll
        for (int q = 0; q < 2; ++q) {         // 512 float4s / 256 threads
            int i  = tid + q * 256;
            int r  = i >> 5;                  // 32 float4 per 128-col row
            int c4 = (i & 31) * 4;
            *(v4f*)(ldsW + r * WSTRIDE + c4) =
                *(const v4f*)(W + (nt * 16 + r) * DIM + c4);
        }
        __syncthreads();

        v8f cu = {};
        v8f ci = {};
        const float* wp = ldsW + m * WSTRIDE + 2 * h;
#pragma unroll
        for (int t = 0; t < 32; ++t) {        // K = 128 in steps of 4
            v2f au = *(const v2f*)(uap + 4 * t);
            v2f ai = *(const v2f*)(iap + 4 * t);
            v2f bw = *(const v2f*)(wp + 4 * t);
            cu = __builtin_amdgcn_wmma_f32_16x16x4_f32(
                     false, au, false, bw, (short)0, cu, false, false);
            ci = __builtin_amdgcn_wmma_f32_16x16x4_f32(
                     false, ai, false, bw, (short)0, ci, false, false);
        }
        float bb = bias[nt * 16 + m];
#pragma unroll
        for (int v = 0; v < 8; ++v)
            partial[v] += (cu[v] + bb) * (ci[v] + bb);
    }

    // reduce over the 16 N-lanes of each half (masks < 16 stay within halves)
#pragma unroll
    for (int v = 0; v < 8; ++v) {
        float p = partial[v];
        p += __shfl_xor(p, 1, 32);
        p += __shfl_xor(p, 2, 32);
        p += __shfl_xor(p, 4, 32);
        p += __shfl_xor(p, 8, 32);
        partial[v] = p;
    }
    if (m == 0) {
#pragma unroll
        for (int v = 0; v < 8; ++v)
            out[row_base + v + 8 * h] = partial[v];
    }
}

extern "C" void kernel_launch(void* const* d_in, const int* in_sizes, int n_in,
                              void* d_out, int out_size, void* d_ws, size_t ws_size,
                              hipStream_t stream) {
    const int*   ui   = (const int*)d_in[0];
    const int*   ii   = (const int*)d_in[1];
    const float* uemb = (const float*)d_in[2];
    const float* iemb = (const float*)d_in[3];
    const float* W    = (const float*)d_in[4];
    const float* bias = (const float*)d_in[5];
    const int*   eidx = (const int*)d_in[6];
    float*       out  = (float*)d_out;

    char* ws = (char*)d_ws;
    int*      slot   = (int*)(ws + WS_SLOT_OFF);
    float*    agg    = (float*)(ws + WS_AGG_OFF);
    unsigned* gcount = (unsigned*)(ws + WS_CNT_OFF);
    int2*     wl     = (int2*)(ws + WS_WL_OFF);

    k_init_slots<<<(NUM_NODES + 255) / 256, 256, 0, stream>>>(slot, gcount);
    k_mark_zero<<<(NSLOTS * 32) / 256, 256, 0, stream>>>(ui, ii, slot, agg);
    k_compact<<<(NUM_EDGES + 255) / 256, 256, 0, stream>>>(eidx, slot, gcount, wl);
    k_apply<<<512, 256, 0, stream>>>(wl, gcount, uemb, iemb, agg);
    k_gemm_dot<<<BATCH / 16 / 8, 256, 0, stream>>>(ui, ii, slot, W, bias, agg, out);
}